// GlobalItemConv_75917841924400
// MI455X (gfx1250) — compile-verified
//
#include <hip/hip_runtime.h>
#include <hip/hip_bf16.h>

// GlobalItemConv on MI455X (gfx1250, wave32).
//
// Pipeline per layer (×3):
//   scatter : h_next[row] += val * h_cur[col]   (f32 HW atomics, 16 thr/edge)
//   norm    : per 16-node tile, one wave computes Gram = H*H^T with 16×
//             v_wmma_f32_16x16x4_f32 (exact f32), takes the diagonal as
//             sum-of-squares, then writes h_cur = h/||h||, out += h/4,
//             and zeroes h_next for the next layer (fused).
//
// d_ws layout: [ h_cur : N*64 f32 | h_next : N*64 f32 ]  (~51.2 MB)

typedef __attribute__((ext_vector_type(2))) float v2f;
typedef __attribute__((ext_vector_type(8))) float v8f;

#define DFEAT 64

__global__ void gic_init(const float* __restrict__ x,
                         float* __restrict__ h_cur,
                         float* __restrict__ h_next,
                         float* __restrict__ out, int n) {
  int i = blockIdx.x * blockDim.x + threadIdx.x;
  if (i < n) {
    float v = x[i];
    h_cur[i]  = v;
    out[i]    = 0.25f * v;   // acc starts at x; final out = acc/4
    h_next[i] = 0.0f;
  }
}

__global__ void gic_scatter(const int* __restrict__ rows,
                            const int* __restrict__ cols,
                            const float* __restrict__ vals,
                            const float* __restrict__ h_cur,
                            float* __restrict__ h_next, int E) {
  int t = blockIdx.x * blockDim.x + threadIdx.x;
  int e = t >> 4;                 // 16 threads per edge
  if (e >= E) return;
  int seg = t & 15;               // 16 segments x 4 floats = 64 feats
  int row = rows[e];
  int col = cols[e];
  float v = vals[e];
  const float4 m = *((const float4*)(h_cur + (size_t)col * DFEAT) + seg);
  float* dst = h_next + (size_t)row * DFEAT + seg * 4;
  // hardware global_atomic_add_f32 (no CAS loop)
  unsafeAtomicAdd(dst + 0, v * m.x);
  unsafeAtomicAdd(dst + 1, v * m.y);
  unsafeAtomicAdd(dst + 2, v * m.z);
  unsafeAtomicAdd(dst + 3, v * m.w);
}

// One wave32 handles one tile of 16 nodes.
__global__ void gic_normalize(float* __restrict__ h_next,
                              float* __restrict__ h_cur,
                              float* __restrict__ out,
                              int N, int n_tiles) {
  int wave = threadIdx.x >> 5;
  int lane = threadIdx.x & 31;
  int tile = blockIdx.x * (blockDim.x >> 5) + wave;
  if (tile >= n_tiles) return;                 // wave-uniform
  int base = tile * 16;
  int rem  = N - base;

  float* hn0 = h_next + (size_t)base * DFEAT;
  float* hc0 = h_cur  + (size_t)base * DFEAT;
  float* o0  = out    + (size_t)base * DFEAT;

  if (rem >= 16) {
    // --- Gram diagonal via V_WMMA_F32_16X16X4_F32 (exact f32) ---
    // A 16x4 f32 layout: lanes 0-15 hold M=lane, K={0,1}; lanes 16-31 hold
    // M=lane-16, K={2,3}. B (4x16) has the mirrored layout, so the SAME
    // registers passed as A and B compute A_chunk * A_chunk^T.
    int rowInTile = lane & 15;
    int khalf     = lane >> 4;                 // 0 or 1
    const float* arow = hn0 + rowInTile * DFEAT + khalf * 2;

    v8f acc = {};
#pragma unroll
    for (int c = 0; c < 16; ++c) {             // 16 chunks of K=4 -> K=64
      v2f a = *(const v2f*)(arow + 4 * c);
      acc = __builtin_amdgcn_wmma_f32_16x16x4_f32(
          /*neg_a=*/false, a, /*neg_b=*/false, a,
          /*c_mod=*/(short)0, acc, /*reuse_a=*/false, /*reuse_b=*/false);
    }

    // --- normalize + accumulate + zero h_next, fully coalesced ---
    // Diagonal (m,m): m<8 -> lane m, acc[m]; m>=8 -> lane m+16, acc[m-8].
#pragma unroll
    for (int r = 0; r < 16; ++r) {
      float ssq = __shfl(acc[r & 7], (r >= 8) ? (r + 16) : r, 32);
      float inv = 1.0f / fmaxf(sqrtf(ssq), 1e-12f);
      float* hp = hn0 + r * DFEAT + lane * 2;
      v2f hv = *(v2f*)hp;
      v2f nv; nv.x = hv.x * inv; nv.y = hv.y * inv;
      *(v2f*)(hc0 + r * DFEAT + lane * 2) = nv;
      float* op = o0 + r * DFEAT + lane * 2;
      v2f ov = *(v2f*)op;
      ov.x += 0.25f * nv.x; ov.y += 0.25f * nv.y;
      *(v2f*)op = ov;
      v2f zz; zz.x = 0.0f; zz.y = 0.0f;
      *(v2f*)hp = zz;                          // reset for next layer
    }
  } else {
    // Tail tile (not hit for N % 16 == 0): one node per lane, scalar path.
    if (lane < rem) {
      float* hp = hn0 + lane * DFEAT;
      float s = 0.0f;
      for (int k = 0; k < DFEAT; ++k) s += hp[k] * hp[k];
      float inv = 1.0f / fmaxf(sqrtf(s), 1e-12f);
      float* hc = hc0 + lane * DFEAT;
      float* op = o0 + lane * DFEAT;
      for (int k = 0; k < DFEAT; ++k) {
        float nv = hp[k] * inv;
        hc[k] = nv;
        op[k] += 0.25f * nv;
        hp[k] = 0.0f;
      }
    }
  }
}

extern "C" void kernel_launch(void* const* d_in, const int* in_sizes, int n_in,
                              void* d_out, int out_size, void* d_ws, size_t ws_size,
                              hipStream_t stream) {
  const float* x    = (const float*)d_in[0];
  const int*   rows = (const int*)d_in[1];
  const int*   cols = (const int*)d_in[2];
  const float* vals = (const float*)d_in[3];
  float*       outp = (float*)d_out;

  const int nfeat = in_sizes[0];        // N * 64
  const int N     = nfeat / DFEAT;
  const int E     = in_sizes[1];

  float* h_cur  = (float*)d_ws;
  float* h_next = h_cur + (size_t)nfeat;

  gic_init<<<(nfeat + 255) / 256, 256, 0, stream>>>(x, h_cur, h_next, outp, nfeat);

  const int n_tiles = (N + 15) / 16;
  const int norm_blocks = (n_tiles + 7) / 8;               // 8 waves / block
  const long scatter_threads = (long)E * 16;
  const int scatter_blocks = (int)((scatter_threads + 255) / 256);

  for (int layer = 0; layer < 3; ++layer) {
    gic_scatter<<<scatter_blocks, 256, 0, stream>>>(rows, cols, vals, h_cur, h_next, E);
    gic_normalize<<<norm_blocks, 256, 0, stream>>>(h_next, h_cur, outp, N, n_tiles);
  }
}